// HistogramLoss_29145648071226
// MI455X (gfx1250) — compile-verified
//
#include <hip/hip_runtime.h>
#include <hip/hip_bf16.h>
#include <math.h>

#define NUM_CLASSES 19
#define BINS_NUM    51
#define DD          256   // feature dims
#define NN          4096  // pixels (64*64)
#define CPAD        32    // padded class stride

typedef __attribute__((ext_vector_type(2))) float v2f;
typedef __attribute__((ext_vector_type(8))) float v8f;

#define TWO_PI 6.283185307179586f
#define LOG2E  1.4426950408889634f

// ---------------------------------------------------------------------------
// Kernel A: per-class stats. One block per feature dim d (256 blocks x 256 thr)
// ---------------------------------------------------------------------------
__global__ void hl_stats(const float* __restrict__ feat, const int* __restrict__ lab,
                         float* __restrict__ sum, float* __restrict__ sum2,
                         float* __restrict__ counts) {
    const int d = blockIdx.x;
    const int t = threadIdx.x;
    __shared__ float s1[NUM_CLASSES], s2[NUM_CLASSES];
    __shared__ unsigned scnt[NUM_CLASSES];
    if (t < NUM_CLASSES) { s1[t] = 0.f; s2[t] = 0.f; scnt[t] = 0u; }
    __syncthreads();
    for (int n = t; n < NN; n += 256) {
        float x = feat[d * NN + n];
        int c = lab[n];
        atomicAdd(&s1[c], x);
        atomicAdd(&s2[c], x * x);
        if (d == 0) atomicAdd(&scnt[c], 1u);
    }
    __syncthreads();
    if (t < NUM_CLASSES) {
        sum[t * DD + d]  = s1[t];
        sum2[t * DD + d] = s2[t];
        if (d == 0) counts[t] = (float)scnt[t];
    }
}

// ---------------------------------------------------------------------------
// Kernel B: finalize stats; write normalized target[c,d,b] and fused per-(d,c)
// sample-KDE constants:
//   A1 = (-0.5/vs) * log2(e)          (exponent slope, exp2 domain)
//   A2 = log2(1/sqrt(2*pi*vs))        (log-prefactor, folded into exponent)
// so the KDE inner loop is K = exp2(fma(df*df, A1, A2)).
// ---------------------------------------------------------------------------
__global__ void hl_finalize(const float* __restrict__ sum, const float* __restrict__ sum2,
                            const float* __restrict__ counts,
                            float* __restrict__ A1, float* __restrict__ A2,
                            float* __restrict__ tgt, float* __restrict__ hasbuf) {
    const int c = blockIdx.x / DD;
    const int d = blockIdx.x % DD;
    const int b = threadIdx.x;   // 0..63
    __shared__ float red[64];

    float cntRaw = counts[c];
    float cnt  = fmaxf(cntRaw, 1.f);
    float miu  = sum[c * DD + d] / cnt;
    float mom2 = sum2[c * DD + d] / cnt;
    float var  = fmaxf(mom2 - miu * miu, 1e-12f);
    float invn = rsqrtf(TWO_PI * var);

    float t = 0.f;
    if (b < BINS_NUM) {
        float binv = -5.f + 0.2f * (float)b;
        float df = binv - miu;
        t = invn * __expf(-0.5f * df * df / var);
    }
    red[b] = t;
    __syncthreads();
    for (int s = 32; s > 0; s >>= 1) {
        if (b < s) red[b] += red[b + s];
        __syncthreads();
    }
    bool has = cntRaw > 0.f;
    float den = has ? fmaxf(red[0], 1e-30f) : 1.f;
    if (b < BINS_NUM)
        tgt[((size_t)c * DD + d) * BINS_NUM + b] = t / den;
    if (b == 0) {
        float vs = var * 0.04f;                       // var / 25
        A1[d * CPAD + c] = (-0.5f / vs) * LOG2E;
        A2[d * CPAD + c] = -0.5f * __log2f(TWO_PI * vs);
        if (d == 0) hasbuf[c] = has ? 1.f : 0.f;
    }
}

// ---------------------------------------------------------------------------
// Kernel C: KDE + segment-sum as one-hot GEMM via V_WMMA_F32_16X16X4_F32.
// One block per d (256 blocks, 128 threads = 4 waves). Wave w owns bins
// [16w, 16w+16). Pixels looped in chunks of 4 (WMMA K=4):
//   A (16x4)  = one-hot(class, pixel)   — two class tiles: 0..15 and 16..31
//   B (4x16)  = K(pixel, bin) = exp2(fma(df^2, q, t)) — 2 exps/lane/chunk
//   D (16x16) = class x bin accumulator
// Branchless: padding bin columns (b>=51) compute with clamped bin value
// (finite garbage, never stored). Per-pixel data packed in LDS as
// float4 {f, q, t, label} -> exactly one ds_load_b128 per pixel.
// ---------------------------------------------------------------------------
__global__ void __launch_bounds__(128)
hl_kde_wmma(const float* __restrict__ feat, const int* __restrict__ lab,
            const float* __restrict__ A1, const float* __restrict__ A2,
            float* __restrict__ samp) {
    const int d    = blockIdx.x;
    const int tid  = threadIdx.x;
    const int wave = tid >> 5;
    const int lane = tid & 31;

    __shared__ float4 s_px[NN];   // 64 KB: {f, q, t, label-bits} per pixel

    for (int i = tid; i < NN; i += 128) {
        int c = lab[i];                       // 19 distinct addrs -> L2 broadcast
        s_px[i] = make_float4(feat[d * NN + i],
                              A1[d * CPAD + c],
                              A2[d * CPAD + c],
                              __int_as_float(c));
    }
    __syncthreads();

    const int col   = lane & 15;
    const int breal = wave * 16 + col;
    const int bcomp = (breal < BINS_NUM) ? breal : (BINS_NUM - 1);  // clamp pad cols
    const float binv = -5.f + 0.2f * (float)bcomp;
    const int khalf = (lane < 16) ? 0 : 2;   // which pixel pair this lane feeds
    const int cls   = col;                   // A-matrix row (class) for tile 0
    const int cls2  = col + 16;              // for tile 1

    v8f acc0 = {}; // classes 0..15
    v8f acc1 = {}; // classes 16..31 (16..18 valid)

    #pragma unroll 4
    for (int nb = 0; nb < NN; nb += 4) {
        const float4 p0 = s_px[nb + khalf];
        const float4 p1 = s_px[nb + khalf + 1];

        const float df0 = binv - p0.x;
        const float df1 = binv - p1.x;
        const float b0 = __builtin_amdgcn_exp2f(fmaf(df0 * df0, p0.y, p0.z));
        const float b1 = __builtin_amdgcn_exp2f(fmaf(df1 * df1, p1.y, p1.z));
        const int c0 = __float_as_int(p0.w);
        const int c1 = __float_as_int(p1.w);

        v2f Bm  = { b0, b1 };
        v2f Am0 = { (c0 == cls)  ? 1.f : 0.f, (c1 == cls)  ? 1.f : 0.f };
        v2f Am1 = { (c0 == cls2) ? 1.f : 0.f, (c1 == cls2) ? 1.f : 0.f };

        acc0 = __builtin_amdgcn_wmma_f32_16x16x4_f32(false, Am0, false, Bm,
                                                     (short)0, acc0, false, false);
        acc1 = __builtin_amdgcn_wmma_f32_16x16x4_f32(false, Am1, false, Bm,
                                                     (short)0, acc1, false, false);
    }

    if (breal < BINS_NUM) {
        const int mofs = (lane < 16) ? 0 : 8;
        #pragma unroll
        for (int v = 0; v < 8; ++v) {
            int c = v + mofs;
            samp[((size_t)c * DD + d) * BINS_NUM + breal] = acc0[v];
            int c2 = c + 16;
            if (c2 < NUM_CLASSES)
                samp[((size_t)c2 * DD + d) * BINS_NUM + breal] = acc1[v];
        }
    }
}

// ---------------------------------------------------------------------------
// Kernel D: per-class smooth-L1 mean. 19 blocks x 256 threads (thread = d).
// ---------------------------------------------------------------------------
__global__ void hl_sl1(const float* __restrict__ samp, const float* __restrict__ tgt,
                       const float* __restrict__ hasbuf, float* __restrict__ sl1sum) {
    const int c = blockIdx.x;
    const int d = threadIdx.x;
    const bool has = hasbuf[c] > 0.5f;
    const float* sp = samp + ((size_t)c * DD + d) * BINS_NUM;
    const float* tp = tgt  + ((size_t)c * DD + d) * BINS_NUM;

    float den = 0.f;
    for (int b = 0; b < BINS_NUM; ++b) den += sp[b];
    den = has ? fmaxf(den, 1e-30f) : 1.f;
    float inv_den = 1.f / den;

    float acc = 0.f;
    for (int b = 0; b < BINS_NUM; ++b) {
        float diff = sp[b] * inv_den - tp[b];
        float ad = fabsf(diff);
        acc += (ad < 1.f) ? 0.5f * diff * diff : (ad - 0.5f);
    }
    __shared__ float red[256];
    red[d] = acc;
    __syncthreads();
    for (int s = 128; s > 0; s >>= 1) {
        if (d < s) red[d] += red[d + s];
        __syncthreads();
    }
    if (d == 0) sl1sum[c] = red[0] / (float)(DD * BINS_NUM);
}

// ---------------------------------------------------------------------------
// Kernel E: final masked mean over classes (wave32 shuffle reduce).
// ---------------------------------------------------------------------------
__global__ void hl_loss(const float* __restrict__ sl1sum, const float* __restrict__ hasbuf,
                        float* __restrict__ out) {
    const int t = threadIdx.x;   // 32 threads = 1 wave
    float v = 0.f, a = 0.f;
    if (t > 0 && t < NUM_CLASSES && hasbuf[t] > 0.5f) { v = sl1sum[t]; a = 1.f; }
    for (int off = 16; off > 0; off >>= 1) {
        v += __shfl_down(v, off);
        a += __shfl_down(a, off);
    }
    if (t == 0) out[0] = v / (a + 1e-12f);   // LOSS_WEIGHT = 1
}

// ---------------------------------------------------------------------------
// Kernel F: pass-through feature copy into out[1..].
// ---------------------------------------------------------------------------
__global__ void hl_copyfeat(const float* __restrict__ feat, float* __restrict__ out, int n) {
    int i = blockIdx.x * blockDim.x + threadIdx.x;
    if (i < n) out[1 + i] = feat[i];
}

// ---------------------------------------------------------------------------
extern "C" void kernel_launch(void* const* d_in, const int* in_sizes, int n_in,
                              void* d_out, int out_size, void* d_ws, size_t ws_size,
                              hipStream_t stream) {
    const float* feat = (const float*)d_in[0];   // [1,256,64,64] -> [D=256][N=4096]
    const int*   lab  = (const int*)d_in[1];     // [4096]
    float* out = (float*)d_out;                  // [0]=loss, [1..]=feature

    // Workspace layout (floats)
    float* ws = (float*)d_ws;
    float* sum    = ws;                                  // 19*256
    float* sum2   = sum    + NUM_CLASSES * DD;           // 19*256
    float* counts = sum2   + NUM_CLASSES * DD;           // 32
    float* hasbuf = counts + 32;                         // 32
    float* A1     = hasbuf + 32;                         // 256*32
    float* A2     = A1     + DD * CPAD;                  // 256*32
    float* tgt    = A2     + DD * CPAD;                  // 19*256*51
    float* samp   = tgt    + (size_t)NUM_CLASSES * DD * BINS_NUM;
    float* sl1sum = samp   + (size_t)NUM_CLASSES * DD * BINS_NUM;  // 32

    hl_stats   <<<DD, 256, 0, stream>>>(feat, lab, sum, sum2, counts);
    hl_finalize<<<NUM_CLASSES * DD, 64, 0, stream>>>(sum, sum2, counts, A1, A2, tgt, hasbuf);
    hl_kde_wmma<<<DD, 128, 0, stream>>>(feat, lab, A1, A2, samp);
    hl_sl1     <<<NUM_CLASSES, 256, 0, stream>>>(samp, tgt, hasbuf, sl1sum);
    hl_loss    <<<1, 32, 0, stream>>>(sl1sum, hasbuf, out);

    const int nfeat = DD * NN;
    hl_copyfeat<<<(nfeat + 255) / 256, 256, 0, stream>>>(feat, out, nfeat);
}